// BaselineRelationalIndependentModel_1030792151185
// MI455X (gfx1250) — compile-verified
//
#include <hip/hip_runtime.h>

// BaselineRelationalIndependentModel: out[i] = sigmoid(W[2048*x[i] + y[i]])
//
// Pure gather + elementwise sigmoid; zero matrix FLOPs (the onehot@W.T in the
// reference is a gather in disguise). Memory-bound:
//   streams: x (64MB) + y (64MB) + out (64MB) coalesced  -> ~8.2us @ 23.3TB/s
//   gather:  W table is 16MB fp32 -> L2-resident (192MB L2)
// Strategy: 128-bit NT loads/stores for the streams (don't evict W from L2),
// default-temporal b32 gathers for W (keep it cached), exp/rcp VALU sigmoid.

typedef int   int4v   __attribute__((ext_vector_type(4)));
typedef float float4v __attribute__((ext_vector_type(4)));

static constexpr int NUM_OBJECTS = 2048;

__global__ __launch_bounds__(256)
void relational_gather_sigmoid_v4(const int* __restrict__ x,
                                  const int* __restrict__ y,
                                  const float* __restrict__ W,
                                  float* __restrict__ out,
                                  int n4) {
  const int i = blockIdx.x * blockDim.x + threadIdx.x;
  if (i >= n4) return;

  // Streaming inputs: non-temporal 128-bit loads (th:TH_LOAD_NT) so the
  // 128MB of index traffic does not evict the 16MB W table from L2.
  const int4v xv = __builtin_nontemporal_load((const int4v*)x + i);
  const int4v yv = __builtin_nontemporal_load((const int4v*)y + i);

  float4v r;
#pragma unroll
  for (int k = 0; k < 4; ++k) {
    const int flat = (xv[k] << 11) + yv[k];   // 2048*x + y, < 4,194,304
    const float w = W[flat];                  // RT gather: L2-resident table
    // sigmoid(w) = 1 / (1 + 2^(-w*log2e)); |w| < 2^-11 so this is sub-ulp.
    const float e = __builtin_amdgcn_exp2f(w * -1.4426950408889634f);
    r[k] = __builtin_amdgcn_rcpf(1.0f + e);
  }

  // Streaming output: non-temporal 128-bit store (th:TH_STORE_NT).
  __builtin_nontemporal_store(r, (float4v*)out + i);
}

__global__ __launch_bounds__(256)
void relational_gather_sigmoid_tail(const int* __restrict__ x,
                                    const int* __restrict__ y,
                                    const float* __restrict__ W,
                                    float* __restrict__ out,
                                    int start, int n) {
  const int i = start + blockIdx.x * blockDim.x + threadIdx.x;
  if (i >= n) return;
  const int flat = (x[i] << 11) + y[i];
  const float w = W[flat];
  const float e = __builtin_amdgcn_exp2f(w * -1.4426950408889634f);
  const float s = __builtin_amdgcn_rcpf(1.0f + e);
  __builtin_nontemporal_store(s, out + i);
}

extern "C" void kernel_launch(void* const* d_in, const int* in_sizes, int n_in,
                              void* d_out, int out_size, void* d_ws, size_t ws_size,
                              hipStream_t stream) {
  const int*   x = (const int*)d_in[0];    // [B] int32
  const int*   y = (const int*)d_in[1];    // [B] int32
  const float* W = (const float*)d_in[2];  // [1, 4194304] fp32
  float*     out = (float*)d_out;          // [B, 1] fp32

  const int n  = in_sizes[0];              // BATCH = 16,777,216
  const int n4 = n >> 2;                   // 4 elements per thread

  if (n4 > 0) {
    const int threads = 256;
    const int blocks  = (n4 + threads - 1) / threads;  // 16384 for BATCH=16M
    relational_gather_sigmoid_v4<<<blocks, threads, 0, stream>>>(x, y, W, out, n4);
  }

  const int done = n4 << 2;
  if (done < n) {
    const int tail = n - done;
    const int threads = 256;
    const int blocks  = (tail + threads - 1) / threads;
    relational_gather_sigmoid_tail<<<blocks, threads, 0, stream>>>(x, y, W, out, done, n);
  }
}